// Net_9878424780941
// MI455X (gfx1250) — compile-verified
//
#include <hip/hip_runtime.h>
#include <math.h>

typedef __attribute__((ext_vector_type(2))) float v2f;
typedef __attribute__((ext_vector_type(8))) float v8f;
typedef __attribute__((ext_vector_type(4))) unsigned int u32x4;
typedef __attribute__((ext_vector_type(4))) int i32x4;
typedef __attribute__((ext_vector_type(8))) int i32x8;

#define F_IN 512
#define HID  16
#define NCLS 40

// ---------------- init: zero accumulators, deg = 1 (self loop) ----------------
__global__ void init_kernel(float* __restrict__ deg, float* __restrict__ out1,
                            float* __restrict__ out, int n) {
  int g = blockIdx.x * blockDim.x + threadIdx.x;
  if (g < n * NCLS) out[g] = 0.0f;
  if (g < n * HID)  out1[g] = 0.0f;
  if (g < n)        deg[g] = 1.0f;
}

// ---------------- degree of source nodes ----------------
__global__ void deg_kernel(const int* __restrict__ ei, float* __restrict__ deg, int E) {
  int e = blockIdx.x * blockDim.x + threadIdx.x;
  if (e < E) atomicAdd(&deg[ei[e]], 1.0f);
}

// ---------------- deg -> deg^-0.5 in place ----------------
__global__ void dinv_kernel(float* __restrict__ deg, int n) {
  int i = blockIdx.x * blockDim.x + threadIdx.x;
  if (i < n) deg[i] = rsqrtf(deg[i]);
}

// ---------------- per-edge norm = dinv[src]*dinv[tgt] ----------------
__global__ void norm_kernel(const int* __restrict__ ei, const float* __restrict__ dinv,
                            float* __restrict__ nrm, int E) {
  int e = blockIdx.x * blockDim.x + threadIdx.x;
  if (e < E) nrm[e] = dinv[ei[e]] * dinv[ei[E + e]];
}

// ---------------- GEMM1: h1 = x @ w1 + b1 via V_WMMA_F32_16X16X4_F32 ----------------
// One wave per 16x16 output tile (N = HID = 16 exactly). K = 512 -> 128 WMMAs/tile.
// w1 (32 KB) is staged into LDS once per block via the Tensor Data Mover; the
// inner loop then issues only the A-stream global_load_b64 plus LDS reads for B.
__global__ void gemm1_wmma(const float* __restrict__ x, const float* __restrict__ w1,
                           const float* __restrict__ b1, float* __restrict__ h1,
                           int ntiles) {
  __shared__ float w1s[F_IN * HID];   // 32 KB

#if __has_builtin(__builtin_amdgcn_tensor_load_to_lds)
  if (threadIdx.x == 0) {             // only wave 0 reaches this block; one TDM op
    unsigned long long ga = (unsigned long long)(uintptr_t)w1;
    unsigned int la = (unsigned int)(uintptr_t)w1s;  // flat low 32 bits = LDS byte offset
    u32x4 g0;
    g0[0] = 1u;                                           // count=1, user descriptor
    g0[1] = la;                                           // lds_addr [63:32]
    g0[2] = (unsigned int)(ga & 0xFFFFFFFFu);             // global_addr lo
    g0[3] = (unsigned int)((ga >> 32) & 0x1FFFFFFu) |     // global_addr hi (bits 56:32)
            (2u << 30);                                   // type=2 ("image")
    i32x8 g1;
    g1[0] = (2 << 16);                  // data_size = 4 bytes
    g1[1] = (int)(8192u << 16);         // tensor_dim0 lo16 at bits [63:48]
    g1[2] = (int)(1u << 16);            // tensor_dim0 hi16 = 0 | tensor_dim1 = 1
    g1[3] = (int)(8192u << 16);         // tensor_dim1 hi = 0 | tile_dim0 = 8192
    g1[4] = 1;                          // tile_dim1 = 1, tile_dim2 = 0
    g1[5] = 8192;                       // tensor_dim0_stride lo32
    g1[6] = 0;
    g1[7] = 0;
    i32x4 g2 = {0, 0, 0, 0};
    i32x4 g3 = {0, 0, 0, 0};
#if __clang_major__ >= 23
    i32x8 g4 = {0, 0, 0, 0, 0, 0, 0, 0};
    __builtin_amdgcn_tensor_load_to_lds(g0, g1, g2, g3, g4, 0);
#else
    __builtin_amdgcn_tensor_load_to_lds(g0, g1, g2, g3, 0);
#endif
    __builtin_amdgcn_s_wait_tensorcnt(0);
  }
  __syncthreads();
#else
  for (int i = threadIdx.x; i < F_IN * HID; i += blockDim.x) w1s[i] = w1[i];
  __syncthreads();
#endif

  int wave = blockIdx.x * (blockDim.x >> 5) + (threadIdx.x >> 5);
  if (wave >= ntiles) return;           // wave-uniform guard; EXEC stays all-ones
  int lane = threadIdx.x & 31;
  int m  = lane & 15;                   // A row / B,C column
  int hi = lane >> 4;
  int kb = hi * 2;                      // K sub-offset per ISA layout

  const float* xa = x + (size_t)(wave * 16 + m) * F_IN + kb;
  v8f c = {};
  for (int k = 0; k < F_IN; k += 4) {
    v2f a = *(const v2f*)(xa + k);      // A[m][k+kb], A[m][k+kb+1]
    v2f b;
    b[0] = w1s[(k + kb) * HID + m];     // B[k+kb][m]   (LDS, conflict-free)
    b[1] = w1s[(k + kb + 1) * HID + m]; // B[k+kb+1][m]
    c = __builtin_amdgcn_wmma_f32_16x16x4_f32(
        /*neg_a=*/false, a, /*neg_b=*/false, b,
        /*c_mod=*/(short)0, c, /*reuse_a=*/false, /*reuse_b=*/false);
  }
  float bias = b1[m];
#pragma unroll
  for (int j = 0; j < 8; ++j) {
    int row = wave * 16 + j + 8 * hi;   // C vgpr j -> row j (+8 for hi half)
    h1[(size_t)row * HID + m] = c[j] + bias;
  }
}

// ---------------- layer-1 edge aggregation: out1[t] += norm_e * h1[s] ----------------
__global__ void agg1_kernel(const int* __restrict__ ei, const float* __restrict__ nrm,
                            const float* __restrict__ h1, float* __restrict__ out1, int E) {
  int g = blockIdx.x * blockDim.x + threadIdx.x;
  if (g >= E * HID) return;
  int e = g >> 4, j = g & 15;
  int s = ei[e], t = ei[E + e];
  atomicAdd(&out1[t * HID + j], nrm[e] * h1[s * HID + j]);
}

// ---------------- self-loop (norm = dinv^2) + ReLU ----------------
__global__ void relu_kernel(const float* __restrict__ dinv, const float* __restrict__ h1,
                            float* __restrict__ out1, int n) {
  int g = blockIdx.x * blockDim.x + threadIdx.x;
  if (g >= n * HID) return;
  float d = dinv[g >> 4];
  out1[g] = fmaxf(out1[g] + d * d * h1[g], 0.0f);
}

// ---------------- GEMM2: h2 = out1 @ w2 + b2 (K=16, tiny; VALU) ----------------
__global__ void gemm2_kernel(const float* __restrict__ out1, const float* __restrict__ w2,
                             const float* __restrict__ b2, float* __restrict__ h2, int n) {
  int g = blockIdx.x * blockDim.x + threadIdx.x;
  if (g >= n * NCLS) return;
  int node = g / NCLS;
  int c = g - node * NCLS;
  float acc = b2[c];
#pragma unroll
  for (int k = 0; k < HID; ++k)
    acc += out1[node * HID + k] * w2[k * NCLS + c];
  h2[g] = acc;
}

// ---------------- layer-2 edge aggregation into d_out ----------------
__global__ void agg2_kernel(const int* __restrict__ ei, const float* __restrict__ nrm,
                            const float* __restrict__ h2, float* __restrict__ out, int E) {
  int g = blockIdx.x * blockDim.x + threadIdx.x;
  if (g >= E * NCLS) return;
  int e = g / NCLS;
  int j = g - e * NCLS;
  int s = ei[e], t = ei[E + e];
  atomicAdd(&out[t * NCLS + j], nrm[e] * h2[s * NCLS + j]);
}

// ---------------- self-loop add + log_softmax, one wave32 per row of 40 ----------------
__global__ void lsm_kernel(const float* __restrict__ dinv, const float* __restrict__ h2,
                           float* __restrict__ out, int n) {
  int row = blockIdx.x * (blockDim.x >> 5) + (threadIdx.x >> 5);
  if (row >= n) return;                 // wave-uniform
  int lane = threadIdx.x & 31;
  float d = dinv[row];
  float d2 = d * d;
  const float* hr = h2 + (size_t)row * NCLS;
  float* orow = out + (size_t)row * NCLS;

  float v0 = orow[lane] + d2 * hr[lane];                 // lanes 0..31 -> ch 0..31
  float v1 = -3.402823466e38f;
  if (lane < NCLS - 32)                                  // lanes 0..7 -> ch 32..39
    v1 = orow[lane + 32] + d2 * hr[lane + 32];

  float mx = fmaxf(v0, v1);
#pragma unroll
  for (int off = 16; off > 0; off >>= 1)
    mx = fmaxf(mx, __shfl_xor(mx, off, 32));
  float s = expf(v0 - mx) + ((lane < NCLS - 32) ? expf(v1 - mx) : 0.0f);
#pragma unroll
  for (int off = 16; off > 0; off >>= 1)
    s += __shfl_xor(s, off, 32);
  float lse = mx + logf(s);

  orow[lane] = v0 - lse;
  if (lane < NCLS - 32) orow[lane + 32] = v1 - lse;
}

extern "C" void kernel_launch(void* const* d_in, const int* in_sizes, int n_in,
                              void* d_out, int out_size, void* d_ws, size_t ws_size,
                              hipStream_t stream) {
  const float* x  = (const float*)d_in[0];
  const int*   ei = (const int*)d_in[1];
  const float* w1 = (const float*)d_in[2];
  const float* b1 = (const float*)d_in[3];
  const float* w2 = (const float*)d_in[4];
  const float* b2 = (const float*)d_in[5];
  float* out = (float*)d_out;

  const int n = in_sizes[0] / F_IN;   // 100000
  const int E = in_sizes[1] / 2;      // 3200000

  // workspace layout (floats): deg/dinv | h1 | out1 | h2 | norm  (~42 MB)
  float* ws   = (float*)d_ws;
  float* deg  = ws;                               // n (becomes dinv in place)
  float* h1   = deg + n;                          // n*HID
  float* out1 = h1 + (size_t)n * HID;             // n*HID
  float* h2   = out1 + (size_t)n * HID;           // n*NCLS
  float* nrm  = h2 + (size_t)n * NCLS;            // E

  const int B = 256;
  init_kernel<<<(n * NCLS + B - 1) / B, B, 0, stream>>>(deg, out1, out, n);
  deg_kernel<<<(E + B - 1) / B, B, 0, stream>>>(ei, deg, E);
  dinv_kernel<<<(n + B - 1) / B, B, 0, stream>>>(deg, n);
  norm_kernel<<<(E + B - 1) / B, B, 0, stream>>>(ei, deg, nrm, E);

  const int ntiles = (n + 15) / 16;               // 6250, exact
  gemm1_wmma<<<(ntiles + 7) / 8, B, 0, stream>>>(x, w1, b1, h1, ntiles);

  long long t1 = (long long)E * HID;
  agg1_kernel<<<(int)((t1 + B - 1) / B), B, 0, stream>>>(ei, nrm, h1, out1, E);
  relu_kernel<<<(n * HID + B - 1) / B, B, 0, stream>>>(deg, h1, out1, n);
  gemm2_kernel<<<(n * NCLS + B - 1) / B, B, 0, stream>>>(out1, w2, b2, h2, n);

  long long t2 = (long long)E * NCLS;
  agg2_kernel<<<(int)((t2 + B - 1) / B), B, 0, stream>>>(ei, nrm, h2, out, E);
  lsm_kernel<<<(n + 7) / 8, B, 0, stream>>>(deg, h2, out, n);
}